// TemporalAggregator_42855183680001
// MI455X (gfx1250) — compile-verified
//
#include <hip/hip_runtime.h>

typedef __attribute__((ext_vector_type(2))) float v2f;
typedef __attribute__((ext_vector_type(8))) float v8f;

// Problem constants (fixed by the reference's setup_inputs)
#define GN 4      // heads / groups
#define BN 2      // batch
#define TN 32     // temporal length (GEMM M and K)
#define CN 128    // channels total
#define CG 32     // channels per group (GEMM N)
#define HN 64     // fine H
#define WN 64     // fine W
#define HC 16     // coarse h
#define WC 16     // coarse w
#define PIX 16    // pixels (w) per workgroup: one full 64B cacheline of floats
#define PP 20     // padded per-(row) pixel stride in LDS floats: 80B -> 16B aligned, bank-friendly
#define NTHREADS 512   // 16 wave32 waves; wave i owns pixel i

__global__ __launch_bounds__(NTHREADS, 1)
void temporal_agg_wmma(const float* __restrict__ x,
                       const float* __restrict__ attn,
                       float* __restrict__ out)
{
    extern __shared__ float smem[];
    float* Xs = smem;               // [k*32+c] slices of PP floats (PIX pixels + pad)
    float* As = smem + 1024 * PP;   // [q*32+k] slices; reused as Out[q*32+c] slices

    const int wblk = blockIdx.x;    // 0..3  (w tile)
    const int h    = blockIdx.y;    // 0..63
    const int gb   = blockIdx.z;    // 0..7  = g*2 + b
    const int gi   = gb >> 1;
    const int bi   = gb & 1;
    const int w0   = wblk * PIX;
    const int tid  = threadIdx.x;

    // ---------- Phase 1a: stage x tile into LDS (coalesced 64B runs) ----------
    // Xs[(k*32+c)*PP + p] = x[bi, k, gi*CG + c, h, w0+p]
    for (int j = tid; j < TN * CG; j += NTHREADS) {
        const int k = j >> 5;
        const int c = j & 31;
        const float* src = x + (((long)bi * TN + k) * CN + (gi * CG + c)) * (long)(HN * WN)
                             + h * WN + w0;
        const float4* s4 = reinterpret_cast<const float4*>(src);
        float4* d4 = reinterpret_cast<float4*>(&Xs[j * PP]);
        d4[0] = s4[0]; d4[1] = s4[1]; d4[2] = s4[2]; d4[3] = s4[3];
    }

    // ---------- Phase 1b: bilinear upsample (half-pixel centers) attn into LDS ----------
    // fine coord -> coarse: y = 0.25*h - 0.375 ; edge taps clamp (== jax renormalized edge)
    const float yy = (float)h * 0.25f - 0.375f;
    const int   y0 = (int)floorf(yy);
    const float fy = yy - (float)y0;
    const int  y0c = min(max(y0, 0), HC - 1);
    const int  y1c = min(max(y0 + 1, 0), HC - 1);

    for (int j = tid; j < TN * TN * PIX; j += NTHREADS) {
        const int p  = j & (PIX - 1);
        const int qk = j >> 4;                    // q*32 + k
        const float xx = (float)(w0 + p) * 0.25f - 0.375f;
        const int   x0 = (int)floorf(xx);
        const float fx = xx - (float)x0;
        const int  x0c = min(max(x0, 0), WC - 1);
        const int  x1c = min(max(x0 + 1, 0), WC - 1);
        const float* ab = attn + ((long)gb * (TN * TN) + qk) * (long)(HC * WC);
        const float a00 = ab[y0c * WC + x0c];
        const float a01 = ab[y0c * WC + x1c];
        const float a10 = ab[y1c * WC + x0c];
        const float a11 = ab[y1c * WC + x1c];
        const float top = a00 + fx * (a01 - a00);
        const float bot = a10 + fx * (a11 - a10);
        As[qk * PP + p] = top + fy * (bot - top);
    }

    __syncthreads();

    // ---------- Phase 2: per-pixel 32x32x32 GEMM via V_WMMA_F32_16X16X4_F32 ----------
    // One wave per pixel. Fragment layout per ISA 7.12.2 (32-bit A 16x4 / B 4x16):
    //   V0 holds K=koff, V1 holds K=koff+1, with koff = 2*(lane>=16); rows/cols = lane&15.
    const int lane = tid & 31;
    const int p    = tid >> 5;           // wave id == pixel
    const int lr   = lane & 15;
    const int koff = (lane >> 4) << 1;

    v8f acc[2][2];
    #pragma unroll
    for (int qt = 0; qt < 2; ++qt)
        #pragma unroll
        for (int ct = 0; ct < 2; ++ct)
            acc[qt][ct] = (v8f){0.f, 0.f, 0.f, 0.f, 0.f, 0.f, 0.f, 0.f};

    #pragma unroll
    for (int ks = 0; ks < 8; ++ks) {
        const int kb = ks * 4 + koff;
        v2f a[2], bm[2];
        #pragma unroll
        for (int qt = 0; qt < 2; ++qt) {
            a[qt].x = As[((16 * qt + lr) * TN + kb    ) * PP + p];
            a[qt].y = As[((16 * qt + lr) * TN + kb + 1) * PP + p];
        }
        #pragma unroll
        for (int ct = 0; ct < 2; ++ct) {
            bm[ct].x = Xs[((kb    ) * CG + 16 * ct + lr) * PP + p];
            bm[ct].y = Xs[((kb + 1) * CG + 16 * ct + lr) * PP + p];
        }
        #pragma unroll
        for (int qt = 0; qt < 2; ++qt)
            #pragma unroll
            for (int ct = 0; ct < 2; ++ct)
                acc[qt][ct] = __builtin_amdgcn_wmma_f32_16x16x4_f32(
                    false, a[qt], false, bm[ct], (short)0, acc[qt][ct], false, false);
    }

    // D layout: VGPR r -> row (r + 8*(lane>=16)), col = lane&15. Write into our own
    // (dead) A slice for this pixel; other waves only touch their own p-slices.
    #pragma unroll
    for (int qt = 0; qt < 2; ++qt)
        #pragma unroll
        for (int ct = 0; ct < 2; ++ct)
            #pragma unroll
            for (int r = 0; r < 8; ++r) {
                const int q = 16 * qt + ((lane >> 4) << 3) + r;
                const int c = 16 * ct + lr;
                As[(q * CG + c) * PP + p] = acc[qt][ct][r];
            }

    __syncthreads();

    // ---------- Phase 3: coalesced 64B stores ----------
    for (int j = tid; j < TN * CG; j += NTHREADS) {
        const int q = j >> 5;
        const int c = j & 31;
        float* dst = out + (((long)bi * TN + q) * CN + (gi * CG + c)) * (long)(HN * WN)
                         + h * WN + w0;
        const float4* s4 = reinterpret_cast<const float4*>(&As[j * PP]);
        float4* d4 = reinterpret_cast<float4*>(dst);
        d4[0] = s4[0]; d4[1] = s4[1]; d4[2] = s4[2]; d4[3] = s4[3];
    }
}

extern "C" void kernel_launch(void* const* d_in, const int* in_sizes, int n_in,
                              void* d_out, int out_size, void* d_ws, size_t ws_size,
                              hipStream_t stream) {
    const float* x    = (const float*)d_in[0];   // (2,32,128,64,64) f32
    const float* attn = (const float*)d_in[1];   // (4,2,32,32,16,16) f32
    float* outp = (float*)d_out;                 // (2,32,128,64,64) f32

    dim3 grid(WN / PIX, HN, GN * BN);            // 4 x 64 x 8 = 2048 workgroups
    dim3 block(NTHREADS);                        // 16 wave32 waves
    const size_t smem = (size_t)2 * 1024 * PP * sizeof(float);   // 160 KB dynamic LDS
    hipLaunchKernelGGL(temporal_agg_wmma, grid, block, smem, stream, x, attn, outp);
}